// SpectralGate_52561809768534
// MI455X (gfx1250) — compile-verified
//
#include <hip/hip_runtime.h>
#include <hip/hip_bf16.h>
#include <math.h>

// ---------------------------------------------------------------------------
// SpectralGate on gfx1250 (MI455X):
//   rfft(32) -> diag(sigmoid(gate)) -> irfft(32)  ==  circulant 32x32 matmul.
//   Memory-bound: 256 MB in + 256 MB out @ 23.3 TB/s ~ 22 us floor.
//   Compute mapped onto V_WMMA_F32_16X16X4_F32 (full f32 precision; needs
//   ~190 TF/s f32 which the WMMA path covers; VALU f32 would not).
//   D is a compile-time constant so every in-loop row offset folds into the
//   signed 24-bit instruction IOFFSET (no per-access 64-bit address math).
// ---------------------------------------------------------------------------

typedef __attribute__((ext_vector_type(2))) float v2f;
typedef __attribute__((ext_vector_type(8))) float v8f;

static constexpr int kD = 2048;            // feature dim (row stride)
static constexpr int kCS = 32;             // chunk size

// --- Kernel 1: build M[32][32] (circulant) from the 17 gate logits ----------
__global__ void sg_build_matrix(const float* __restrict__ fg,
                                float* __restrict__ Mbuf) {
    int idx = blockIdx.x * blockDim.x + threadIdx.x;   // 0..1023
    if (idx >= 32 * 32) return;
    int i = idx >> 5;
    int j = idx & 31;
    int d = (i - j) & 31;

    float s0  = 1.0f / (1.0f + __expf(-fg[0]));
    float s16 = 1.0f / (1.0f + __expf(-fg[16]));
    float acc = s0;
    const float w = 6.283185307179586476925f / 32.0f;
    #pragma clang loop unroll(disable)
    for (int k = 1; k <= 15; ++k) {
        float sk = 1.0f / (1.0f + __expf(-fg[k]));
        acc += 2.0f * sk * cosf(w * (float)(k * d));
    }
    acc += s16 * ((d & 1) ? -1.0f : 1.0f);
    Mbuf[idx] = acc * (1.0f / 32.0f);
}

// --- Kernel 2: out_chunk(32xD) = M(32x32) * x_chunk(32xD) via f32 WMMA ------
// Grid: (nChunks, 2). Block: 256 threads = 8 wave32 waves. Wave w of block
// (cx, cy) owns column tiles  w + 8*cy, +16, ...  (kD/16 = 128 tiles -> 8
// uniform iterations; EXEC stays all-ones as WMMA requires).
__global__ __launch_bounds__(256) void sg_apply_wmma(
    const float* __restrict__ x, const float* __restrict__ Mbuf,
    float* __restrict__ out) {

    const int chunk = blockIdx.x;
    const int lane  = threadIdx.x & 31;
    const int wave  = threadIdx.x >> 5;   // 0..7
    const int half  = lane >> 4;          // half-wave select (K split)
    const int n     = lane & 15;          // row (A) / column (B, C, D)

    // Preload A fragments: 2 row-tiles x 8 K-steps of the 32x32 matrix.
    // A 16x4 f32 layout: lane m (0-15): V0=A[m][k0],   V1=A[m][k0+1];
    //                    lane m+16   : V0=A[m][k0+2], V1=A[m][k0+3].
    v2f a[2][8];
    #pragma unroll
    for (int r = 0; r < 2; ++r) {
        const int row = r * 16 + n;
        #pragma unroll
        for (int ks = 0; ks < 8; ++ks) {
            const int k = ks * 4 + 2 * half;
            a[r][ks].x = Mbuf[row * 32 + k];
            a[r][ks].y = Mbuf[row * 32 + k + 1];
        }
    }

    const long long chunkBase = (long long)chunk * (kCS * kD);
    // Per-lane bases: fold ALL lane-varying row terms in; with kD constexpr,
    // every in-loop offset below is a compile-time immediate (< 2^23 bytes).
    const float* __restrict__ xbase = x   + chunkBase + (2 * half) * kD + n;
    float*       __restrict__ obase = out + chunkBase + (8 * half) * kD + n;

    const int nTiles = kD >> 4;           // 128
    const int start  = wave + 8 * blockIdx.y;

    for (int tile = start; tile < nTiles; tile += 16) {
        const int coff = tile * 16;       // uniform per iteration
        const float* __restrict__ xt = xbase + coff;
        float*       __restrict__ ot = obase + coff;

        v8f c0 = {};   // rows 0..15 of the output tile
        v8f c1 = {};   // rows 16..31

        #pragma unroll
        for (int ks = 0; ks < 8; ++ks) {
            // B 4x16 f32 layout: lane n: V0=B[k0][n], V1=B[k0+1][n];
            //                    lane n+16: V0=B[k0+2][n], V1=B[k0+3][n].
            // xt already contains the +2*half rows; offsets are immediates.
            v2f b;
            b.x = xt[ks * 4 * kD];
            b.y = xt[ks * 4 * kD + kD];

            c0 = __builtin_amdgcn_wmma_f32_16x16x4_f32(
                     false, a[0][ks], false, b, (short)0, c0, false, false);
            c1 = __builtin_amdgcn_wmma_f32_16x16x4_f32(
                     false, a[1][ks], false, b, (short)0, c1, false, false);
        }

        // D 16x16 f32 layout: VGPR r: lanes 0-15 -> M=r, lanes 16-31 -> M=8+r.
        // obase already contains the +8*half rows; offsets are immediates.
        #pragma unroll
        for (int r = 0; r < 8; ++r) {
            ot[r * kD]        = c0[r];
            ot[(16 + r) * kD] = c1[r];
        }
    }
}

extern "C" void kernel_launch(void* const* d_in, const int* in_sizes, int n_in,
                              void* d_out, int out_size, void* d_ws, size_t ws_size,
                              hipStream_t stream) {
    const float* x  = (const float*)d_in[0];   // (8, 4096, 2048) f32
    const float* fg = (const float*)d_in[1];   // (17,) f32
    float* out  = (float*)d_out;
    float* Mbuf = (float*)d_ws;                // 32*32*4 = 4 KB scratch

    const int total = in_sizes[0];             // B*T*D
    const int nChunks = total / (kCS * kD);    // 1024

    sg_build_matrix<<<4, 256, 0, stream>>>(fg, Mbuf);
    dim3 grid(nChunks, 2, 1);
    sg_apply_wmma<<<grid, 256, 0, stream>>>(x, Mbuf, out);
}